// Generator3DLUT_identity_28467043237891
// MI455X (gfx1250) — compile-verified
//
#include <hip/hip_runtime.h>

// ---------------- problem constants ----------------
#define LUT_D   33
#define NLUT    (LUT_D * LUT_D * LUT_D)       // 35937
#define IMG_HW  (1024 * 1024)
#define N_IMG   8
#define TILE_PIX 1024                          // pixels per block tile
#define THREADS  256                           // 8 waves (wave32)

typedef unsigned int       u32;
typedef unsigned long long u64;
typedef u32 u32x4 __attribute__((ext_vector_type(4)));
typedef int i32x4 __attribute__((ext_vector_type(4)));
typedef int i32x8 __attribute__((ext_vector_type(8)));

#if defined(__has_builtin)
#if __has_builtin(__builtin_amdgcn_tensor_load_to_lds) && __has_builtin(__builtin_amdgcn_s_wait_tensorcnt)
#define USE_TDM 1
#endif
#endif
#ifndef USE_TDM
#define USE_TDM 0
#endif

// therock-10.0 headers ship the TDM descriptor header and use the 6-arg builtin;
// ROCm 7.2 (clang-22) uses the 5-arg form.
#if defined(__has_include)
#if __has_include(<hip/amd_detail/amd_gfx1250_TDM.h>)
#define TDM_6ARG 1
#endif
#endif
#ifndef TDM_6ARG
#define TDM_6ARG 0
#endif

// ---------------- LUT repack: (3, 33^3) planar -> (33^3) float4 ----------------
__global__ void lut_pack_kernel(const float* __restrict__ lut, float4* __restrict__ lut4) {
    int t = blockIdx.x * blockDim.x + threadIdx.x;
    if (t < NLUT) {
        lut4[t] = make_float4(lut[t], lut[t + NLUT], lut[t + 2 * NLUT], 0.0f);
    }
}

__device__ __forceinline__ float lerpf(float a, float b, float w) {
    return fmaf(w, b - a, a);
}

// indices: ch0 -> k (fastest), ch1 -> j, ch2 -> i ; idx = (i*33 + j)*33 + k
__device__ __forceinline__ void coords(float r, float g, float b,
                                       int& base00, int& b01, int& b10, int& b11,
                                       int& k0, int& k1,
                                       float& wk, float& wj, float& wi) {
    float tr = fminf(fmaxf(r * 32.0f, 0.0f), 32.0f);
    float tg = fminf(fmaxf(g * 32.0f, 0.0f), 32.0f);
    float tb = fminf(fmaxf(b * 32.0f, 0.0f), 32.0f);
    k0 = (int)tr; int j0 = (int)tg; int i0 = (int)tb;
    wk = tr - (float)k0; wj = tg - (float)j0; wi = tb - (float)i0;
    k1 = (k0 < 32) ? k0 + 1 : 32;
    int jd = (j0 < 32) ? LUT_D : 0;                 // +1 row in j
    int id = (i0 < 32) ? LUT_D * LUT_D : 0;         // +1 slab in i
    base00 = (i0 * LUT_D + j0) * LUT_D;
    b01 = base00 + jd;
    b10 = base00 + id;
    b11 = base00 + id + jd;
}

__device__ __forceinline__ void apply_packed(const float4* __restrict__ P,
                                             float r, float g, float b,
                                             float& o0, float& o1, float& o2) {
    int base00, b01, b10, b11, k0, k1; float wk, wj, wi;
    coords(r, g, b, base00, b01, b10, b11, k0, k1, wk, wj, wi);
    float4 v000 = P[base00 + k0], v001 = P[base00 + k1];
    float4 v010 = P[b01 + k0],    v011 = P[b01 + k1];
    float4 v100 = P[b10 + k0],    v101 = P[b10 + k1];
    float4 v110 = P[b11 + k0],    v111 = P[b11 + k1];
    float c00x = lerpf(v000.x, v001.x, wk), c00y = lerpf(v000.y, v001.y, wk), c00z = lerpf(v000.z, v001.z, wk);
    float c01x = lerpf(v010.x, v011.x, wk), c01y = lerpf(v010.y, v011.y, wk), c01z = lerpf(v010.z, v011.z, wk);
    float c10x = lerpf(v100.x, v101.x, wk), c10y = lerpf(v100.y, v101.y, wk), c10z = lerpf(v100.z, v101.z, wk);
    float c11x = lerpf(v110.x, v111.x, wk), c11y = lerpf(v110.y, v111.y, wk), c11z = lerpf(v110.z, v111.z, wk);
    float c0x = lerpf(c00x, c01x, wj), c0y = lerpf(c00y, c01y, wj), c0z = lerpf(c00z, c01z, wj);
    float c1x = lerpf(c10x, c11x, wj), c1y = lerpf(c10y, c11y, wj), c1z = lerpf(c10z, c11z, wj);
    o0 = lerpf(c0x, c1x, wi);
    o1 = lerpf(c0y, c1y, wi);
    o2 = lerpf(c0z, c1z, wi);
}

__device__ __forceinline__ void apply_planar(const float* __restrict__ L,
                                             float r, float g, float b,
                                             float& o0, float& o1, float& o2) {
    int base00, b01, b10, b11, k0, k1; float wk, wj, wi;
    coords(r, g, b, base00, b01, b10, b11, k0, k1, wk, wj, wi);
    float oc[3];
#pragma unroll
    for (int c = 0; c < 3; ++c) {
        const float* Lc = L + c * NLUT;
        float v000 = Lc[base00 + k0], v001 = Lc[base00 + k1];
        float v010 = Lc[b01 + k0],    v011 = Lc[b01 + k1];
        float v100 = Lc[b10 + k0],    v101 = Lc[b10 + k1];
        float v110 = Lc[b11 + k0],    v111 = Lc[b11 + k1];
        float c00 = lerpf(v000, v001, wk), c01 = lerpf(v010, v011, wk);
        float c10 = lerpf(v100, v101, wk), c11 = lerpf(v110, v111, wk);
        oc[c] = lerpf(lerpf(c00, c01, wj), lerpf(c10, c11, wj), wi);
    }
    o0 = oc[0]; o1 = oc[1]; o2 = oc[2];
}

// ---------------- main kernel ----------------
__global__ __launch_bounds__(THREADS) void lut3d_apply_kernel(
    const float* __restrict__ x, const float* __restrict__ lutp,
    const float4* __restrict__ lut4, float* __restrict__ out) {
    __shared__ float tile[3 * TILE_PIX];   // 12 KB: 3 channel rows of the pixel tile

    const int tilesPerImg = IMG_HW / TILE_PIX;                    // 1024
    const int img    = blockIdx.x / tilesPerImg;
    const int tstart = (blockIdx.x % tilesPerImg) * TILE_PIX;
    const size_t base = (size_t)img * 3 * IMG_HW + (size_t)tstart; // channel-0 tile start

#if USE_TDM
    // TDM: one 2D DMA per block -> LDS. tile = 1024 floats x 3 rows, row stride = IMG_HW floats.
    if (threadIdx.x < 32) {                 // wave 0 issues the DMA (TDM ignores EXEC)
        u64 gaddr   = (u64)(const void*)(x + base);
        u32 lds_off = (u32)(size_t)(&tile[0]);   // generic LDS addr low 32 bits == LDS byte offset
        u64 stride0 = (u64)IMG_HW;               // dim0 stride in data_size units

        u32x4 g0;
        g0[0] = 1u;                                            // count=1, is_restore=0, gather off
        g0[1] = lds_off;                                       // lds_addr
        g0[2] = (u32)(gaddr & 0xFFFFFFFFu);                    // global_addr[31:0]
        g0[3] = (u32)((gaddr >> 32) & 0x1FFFFFFu) | (2u << 30); // global_addr[56:32] | type=2

        i32x8 g1;
        g1[0] = (int)(2u << 16);                               // data_size=2 (4 bytes), no pad/iter
        g1[1] = (int)(((u32)TILE_PIX & 0xFFFFu) << 16);        // tensor_dim0[15:0]
        g1[2] = (int)((((u32)TILE_PIX >> 16) & 0xFFFFu) | (3u << 16)); // tensor_dim0 hi | tensor_dim1 lo
        g1[3] = (int)(((u32)TILE_PIX & 0xFFFFu) << 16);        // tensor_dim1 hi (0) | tile_dim0
        g1[4] = (int)3;                                        // tile_dim1 = 3, tile_dim2 = 0
        g1[5] = (int)(u32)(stride0 & 0xFFFFFFFFu);             // tensor_dim0_stride[31:0]
        g1[6] = (int)(u32)((stride0 >> 32) & 0xFFFFu);         // stride0[47:32] | stride1 lo (0)
        g1[7] = 0;                                             // stride1 hi
        i32x4 gz4 = {0, 0, 0, 0};
#if TDM_6ARG
        i32x8 gz8 = {0, 0, 0, 0, 0, 0, 0, 0};
        __builtin_amdgcn_tensor_load_to_lds(g0, g1, gz4, gz4, gz8, 0);
#else
        __builtin_amdgcn_tensor_load_to_lds(g0, g1, gz4, gz4, 0);
#endif
        __builtin_amdgcn_s_wait_tensorcnt(0);
    }
    __syncthreads();
#else
    {   // cooperative B128 staging fallback
        float4* tp = (float4*)tile;
        tp[threadIdx.x]                        = ((const float4*)(x + base))[threadIdx.x];
        tp[(TILE_PIX / 4) + threadIdx.x]       = ((const float4*)(x + base + IMG_HW))[threadIdx.x];
        tp[2 * (TILE_PIX / 4) + threadIdx.x]   = ((const float4*)(x + base + 2 * IMG_HW))[threadIdx.x];
    }
    __syncthreads();
#endif

    const int t = threadIdx.x;
    const float4* tp = (const float4*)tile;
    float4 Rv = tp[t];
    float4 Gv = tp[(TILE_PIX / 4) + t];
    float4 Bv = tp[2 * (TILE_PIX / 4) + t];

    float rA[4] = {Rv.x, Rv.y, Rv.z, Rv.w};
    float gA[4] = {Gv.x, Gv.y, Gv.z, Gv.w};
    float bA[4] = {Bv.x, Bv.y, Bv.z, Bv.w};
    float o0[4], o1[4], o2[4];

    if (lut4) {
#pragma unroll
        for (int p = 0; p < 4; ++p)
            apply_packed(lut4, rA[p], gA[p], bA[p], o0[p], o1[p], o2[p]);
    } else {
#pragma unroll
        for (int p = 0; p < 4; ++p)
            apply_planar(lutp, rA[p], gA[p], bA[p], o0[p], o1[p], o2[p]);
    }

    ((float4*)(out + base))[t]              = make_float4(o0[0], o0[1], o0[2], o0[3]);
    ((float4*)(out + base + IMG_HW))[t]     = make_float4(o1[0], o1[1], o1[2], o1[3]);
    ((float4*)(out + base + 2 * IMG_HW))[t] = make_float4(o2[0], o2[1], o2[2], o2[3]);
}

// ---------------- launcher ----------------
extern "C" void kernel_launch(void* const* d_in, const int* in_sizes, int n_in,
                              void* d_out, int out_size, void* d_ws, size_t ws_size,
                              hipStream_t stream) {
    const float* x   = (const float*)d_in[0];   // (8, 3, 1024, 1024) f32
    const float* lut = (const float*)d_in[1];   // (3, 33, 33, 33)    f32
    float* out = (float*)d_out;                 // (8, 3, 1024, 1024) f32
    (void)in_sizes; (void)n_in; (void)out_size;

    const size_t packed_bytes = (size_t)NLUT * sizeof(float4);
    const bool use_packed = (d_ws != nullptr) && (ws_size >= packed_bytes);

    const float4* lut4 = nullptr;
    if (use_packed) {
        lut_pack_kernel<<<(NLUT + 255) / 256, 256, 0, stream>>>(lut, (float4*)d_ws);
        lut4 = (const float4*)d_ws;
    }

    const int blocks = N_IMG * (IMG_HW / TILE_PIX);   // 8192
    lut3d_apply_kernel<<<blocks, THREADS, 0, stream>>>(x, lut, lut4, out);
}